// HeteroSAGE_38628935860965
// MI455X (gfx1250) — compile-verified
//
#include <hip/hip_runtime.h>
#include <hip/hip_bf16.h>

typedef float v2f __attribute__((ext_vector_type(2)));
typedef float v8f __attribute__((ext_vector_type(8)));

#define N_NODES 100000
#define N_BLK   (N_NODES / 16)   // 6250, exact
#define EDGES   1600000
#define K2DIM   128              // K = 256 packed as 128 pairs

// ---------- Pack combined weight W = [wl ; wr] (256 x ncols) into WMMA pair
// layout: Wp[k2][n][2] = { W[2k2][n], W[2k2+1][n] } ----------
__global__ void pack_w(const float* __restrict__ wl, const float* __restrict__ wr,
                       float* __restrict__ wp, int ncols) {
    int idx = blockIdx.x * blockDim.x + threadIdx.x;   // over 256*ncols
    if (idx >= 256 * ncols) return;
    int k = idx / ncols, n = idx - k * ncols;
    float v = (k < 128) ? wl[k * ncols + n] : wr[(k - 128) * ncols + n];
    wp[((k >> 1) * ncols + n) * 2 + (k & 1)] = v;
}

// ---------- Edge scatter: one wave per edge, lane owns a float4 slice.
// agg[dst] += ew * xsrc[src]; cnt[dst] += 1 (lane 0). ----------
__global__ void scatter_add(const int* __restrict__ src, const int* __restrict__ dst,
                            const float* __restrict__ ew, const float* __restrict__ xsrc,
                            float* __restrict__ agg, float* __restrict__ cnt, int nE) {
    int t = blockIdx.x * blockDim.x + threadIdx.x;
    int e = t >> 5;
    int lane = t & 31;
    if (e >= nE) return;
    int s = src[e];
    int d = dst[e];
    float w = ew[e];
    const float4* xs = reinterpret_cast<const float4*>(xsrc) + (size_t)s * 32;
    float4 v = xs[lane];
    float* ap = agg + (size_t)d * 128 + lane * 4;
    unsafeAtomicAdd(ap + 0, v.x * w);
    unsafeAtomicAdd(ap + 1, v.y * w);
    unsafeAtomicAdd(ap + 2, v.z * w);
    unsafeAtomicAdd(ap + 3, v.w * w);
    if (cnt != nullptr && lane == 0) unsafeAtomicAdd(cnt + d, 1.0f);
}

// ---------- Build A = [mean ; x_dst] (N x 256) in WMMA pair layout:
// Ap[nb][k2][m][2], nb=node/16, m=node%16 ----------
__global__ void pack_a(const float* __restrict__ agg, const float* __restrict__ cnt,
                       const float* __restrict__ xdst, float* __restrict__ ap) {
    int idx = blockIdx.x * blockDim.x + threadIdx.x;   // over N_NODES*256
    int node = idx >> 8;
    int k = idx & 255;
    if (node >= N_NODES) return;
    float v;
    if (k < 128) {
        float c = cnt[node];
        v = agg[(size_t)node * 128 + k] / fmaxf(c, 1.0f);
    } else {
        v = xdst[(size_t)node * 128 + (k - 128)];
    }
    int nb = node >> 4, m = node & 15;
    ap[(((size_t)nb * K2DIM + (k >> 1)) * 16 + m) * 2 + (k & 1)] = v;
}

// ---------- Hidden GEMM: h = relu(A @ W + b), M-tile=16 nodes/block,
// 8 waves each own a 16-column N-tile, K=256 via 64 x v_wmma_f32_16x16x4_f32 ----------
__global__ void gemm_h(const float* __restrict__ ap, const float* __restrict__ wp,
                       const float* __restrict__ bias, float* __restrict__ h) {
    int lane = threadIdx.x & 31;
    int wv = threadIdx.x >> 5;          // 0..7 -> N-tile
    int nb = blockIdx.x;                // node block
    int hi = lane >> 4;
    int nl = lane & 15;
    int n = wv * 16 + nl;               // output column
    float b0 = bias[n];
    v8f acc;
#pragma unroll
    for (int i = 0; i < 8; ++i) acc[i] = b0;   // bias folded into C
    const float* apb = ap + (size_t)nb * K2DIM * 32;
#pragma unroll 4
    for (int s = 0; s < 64; ++s) {
        int k2 = 2 * s + hi;                                    // lanes 16-31: K+2/K+3
        v2f a = *(const v2f*)(apb + (k2 * 16 + nl) * 2);        // A[m][k],A[m][k+1]
        v2f b = *(const v2f*)(wp + ((size_t)k2 * 128 + n) * 2); // W[k][n],W[k+1][n]
        acc = __builtin_amdgcn_wmma_f32_16x16x4_f32(false, a, false, b,
                                                    (short)0, acc, false, false);
    }
    float* hb = h + (size_t)nb * 16 * 128;
#pragma unroll
    for (int i = 0; i < 8; ++i) {
        int row = i + 8 * hi;
        hb[row * 128 + n] = fmaxf(acc[i], 0.0f);
    }
}

// ---------- Output GEMM (C=16) + log_softmax. One wave per 16-node block. ----------
__global__ void gemm_out(const float* __restrict__ ap, const float* __restrict__ wp,
                         const float* __restrict__ bias, float* __restrict__ out) {
    int lane = threadIdx.x & 31;
    int wv = threadIdx.x >> 5;
    int nb = blockIdx.x * 8 + wv;
    if (nb >= N_BLK) return;            // wave-uniform, EXEC stays all-ones for WMMA
    int hi = lane >> 4;
    int nl = lane & 15;
    float b0 = bias[nl];
    v8f acc;
#pragma unroll
    for (int i = 0; i < 8; ++i) acc[i] = b0;
    const float* apb = ap + (size_t)nb * K2DIM * 32;
#pragma unroll 4
    for (int s = 0; s < 64; ++s) {
        int k2 = 2 * s + hi;
        v2f a = *(const v2f*)(apb + (k2 * 16 + nl) * 2);
        v2f b = *(const v2f*)(wp + (k2 * 16 + nl) * 2);
        acc = __builtin_amdgcn_wmma_f32_16x16x4_f32(false, a, false, b,
                                                    (short)0, acc, false, false);
    }
    float* ob = out + (size_t)nb * 16 * 16;
#pragma unroll
    for (int i = 0; i < 8; ++i) {
        // row m = i + 8*hi lives across the 16 lanes of this half-wave
        float v = acc[i];
        float mx = v;
        mx = fmaxf(mx, __shfl_xor(mx, 1));
        mx = fmaxf(mx, __shfl_xor(mx, 2));
        mx = fmaxf(mx, __shfl_xor(mx, 4));
        mx = fmaxf(mx, __shfl_xor(mx, 8));
        float ex = expf(v - mx);
        float sum = ex;
        sum += __shfl_xor(sum, 1);
        sum += __shfl_xor(sum, 2);
        sum += __shfl_xor(sum, 4);
        sum += __shfl_xor(sum, 8);
        int row = i + 8 * hi;
        ob[row * 16 + nl] = (v - mx) - logf(sum);
    }
}

extern "C" void kernel_launch(void* const* d_in, const int* in_sizes, int n_in,
                              void* d_out, int out_size, void* d_ws, size_t ws_size,
                              hipStream_t stream) {
    const float* x_user  = (const float*)d_in[0];
    const float* x_item  = (const float*)d_in[1];
    const int*   ei_u2i  = (const int*)d_in[2];
    const int*   ei_i2u  = (const int*)d_in[3];
    const float* ew_u2i  = (const float*)d_in[4];
    const float* ew_i2u  = (const float*)d_in[5];
    const float* w1l_u2i = (const float*)d_in[6];
    const float* b1l_u2i = (const float*)d_in[7];
    const float* w1r_u2i = (const float*)d_in[8];
    const float* w1l_i2u = (const float*)d_in[9];
    const float* b1l_i2u = (const float*)d_in[10];
    const float* w1r_i2u = (const float*)d_in[11];
    const float* w2l     = (const float*)d_in[12];
    const float* b2l     = (const float*)d_in[13];
    const float* w2r     = (const float*)d_in[14];
    float* out = (float*)d_out;

    float* ws = (float*)d_ws;
    size_t o = 0;
    float* Wp_u2i = ws + o; o += 32768;                   // 128 k2 * 128 n * 2
    float* Wp_i2u = ws + o; o += 32768;
    float* Wp2    = ws + o; o += 4096;                    // 128 k2 * 16 n * 2
    float* agg_a  = ws + o; o += (size_t)N_NODES * 128;   // agg_item, later agg2
    float* agg_b  = ws + o; o += (size_t)N_NODES * 128;   // agg_user
    float* cnt_i  = ws + o; o += N_NODES;
    float* cnt_u  = ws + o; o += N_NODES;
    float* Ap     = ws + o; o += (size_t)N_NODES * 256;   // packed [mean;x]
    float* h_item = ws + o; o += (size_t)N_NODES * 128;
    float* h_user = ws + o; o += (size_t)N_NODES * 128;

    // ---- pack weights into WMMA pair layout ----
    pack_w<<<128, 256, 0, stream>>>(w1l_u2i, w1r_u2i, Wp_u2i, 128);
    pack_w<<<128, 256, 0, stream>>>(w1l_i2u, w1r_i2u, Wp_i2u, 128);
    pack_w<<<16,  256, 0, stream>>>(w2l,     w2r,     Wp2,    16);

    const int sgrid = (EDGES * 32) / 256;   // 200000 blocks, 8 edges/block

    // ---- layer 1: user -> item ----
    hipMemsetAsync(agg_a, 0, (size_t)N_NODES * 128 * sizeof(float), stream);
    hipMemsetAsync(cnt_i, 0, N_NODES * sizeof(float), stream);
    scatter_add<<<sgrid, 256, 0, stream>>>(ei_u2i, ei_u2i + EDGES, ew_u2i,
                                           x_user, agg_a, cnt_i, EDGES);
    // ---- layer 1: item -> user ----
    hipMemsetAsync(agg_b, 0, (size_t)N_NODES * 128 * sizeof(float), stream);
    hipMemsetAsync(cnt_u, 0, N_NODES * sizeof(float), stream);
    scatter_add<<<sgrid, 256, 0, stream>>>(ei_i2u, ei_i2u + EDGES, ew_i2u,
                                           x_item, agg_b, cnt_u, EDGES);

    pack_a<<<N_NODES, 256, 0, stream>>>(agg_a, cnt_i, x_item, Ap);
    gemm_h<<<N_BLK, 256, 0, stream>>>(Ap, Wp_u2i, b1l_u2i, h_item);
    pack_a<<<N_NODES, 256, 0, stream>>>(agg_b, cnt_u, x_user, Ap);
    gemm_h<<<N_BLK, 256, 0, stream>>>(Ap, Wp_i2u, b1l_i2u, h_user);

    // ---- output layer: item -> user on h_item (reuse cnt_u: same dst indices) ----
    hipMemsetAsync(agg_a, 0, (size_t)N_NODES * 128 * sizeof(float), stream);
    scatter_add<<<sgrid, 256, 0, stream>>>(ei_i2u, ei_i2u + EDGES, ew_i2u,
                                           h_item, agg_a, nullptr, EDGES);
    pack_a<<<N_NODES, 256, 0, stream>>>(agg_a, cnt_u, h_user, Ap);
    gemm_out<<<(N_BLK + 7) / 8, 256, 0, stream>>>(Ap, Wp2, b2l, out);
}